// Model_670014898618
// MI455X (gfx1250) — compile-verified
//
#include <hip/hip_runtime.h>
#include <hip/hip_bf16.h>
#include <math.h>

typedef __attribute__((ext_vector_type(2))) float v2f;
typedef __attribute__((ext_vector_type(8))) float v8f;

#define BATCH 32
#define SEQ   1024
#define DIM   256
#define L1    1030      // 1024 + 2*6 - 7 + 1
#define KP1   515       // max(4, ceil(1030/2))
#define CH1   768       // 128 groups * 6
#define L2    519       // 515 + 2*4 - 5 + 1
#define CH2   896       // 64 groups * 14
#define NCLS  1000
#define KFLAT 3584      // 896 * 4

// ---------------------------------------------------------------------------
// Kernel 1: fused embedding gather + depthwise conv1 (K=7, pad=6) + fold-sum
// One block per (batch b, folded group j in [0,128)). Input channels 2j,2j+1
// are adjacent -> single 8-byte gather per position.
// Writes h1[b, j*6+f, l], l in [0,1030).
// ---------------------------------------------------------------------------
__global__ __launch_bounds__(256)
void k1_embed_conv_fold(const int* __restrict__ x, const float* __restrict__ emb,
                        const float* __restrict__ w1, const float* __restrict__ b1,
                        float* __restrict__ h1)
{
    __shared__ float e0[SEQ + 12];
    __shared__ float e1[SEQ + 12];
    __shared__ float wl[84];
    const int tid = threadIdx.x;
    const int b   = blockIdx.x >> 7;
    const int j   = blockIdx.x & 127;

    __builtin_prefetch(w1 + 84 * j, 0, 0);   // global_prefetch_b8

    if (tid < 6) { e0[tid] = 0.f; e1[tid] = 0.f; e0[SEQ + 6 + tid] = 0.f; e1[SEQ + 6 + tid] = 0.f; }
    if (tid < 84) wl[tid] = w1[84 * j + tid];   // rows 12j..12j+11 contiguous

    const int d0 = 2 * j;                       // even -> 8B-aligned pair (d0, d0+1)
    for (int s = tid; s < SEQ; s += 256) {
        int tok = x[b * SEQ + s];
        v2f ev = *(const v2f*)(emb + (size_t)tok * DIM + d0);  // global_load_b64 gather
        e0[6 + s] = ev.x;
        e1[6 + s] = ev.y;
    }
    __syncthreads();

    float* outBase = h1 + (size_t)(b * 128 + j) * 6 * L1;
    for (int idx = tid; idx < 6 * L1; idx += 256) {
        int f = idx / L1, l = idx - f * L1;
        const float* wa = wl + f * 7;
        const float* wb = wl + 42 + f * 7;
        float acc = b1[12 * j + f] + b1[12 * j + 6 + f];
        #pragma unroll
        for (int k = 0; k < 7; ++k)
            acc += e0[l + k] * wa[k] + e1[l + k] * wb[k];
        outBase[(size_t)f * L1 + l] = acc;
    }
}

// ---------------------------------------------------------------------------
// Kernel 2: exact order-preserving k-max pool (k=515 of 1030) + tanh.
// One block per (b, channel) row. Threshold = 515th-largest key via 4-pass
// MSB radix select (256-bin LDS histogram per pass); ties kept lowest-index-
// first (matches lax.top_k); positions via block-level order-preserving scan.
// ---------------------------------------------------------------------------
__device__ __forceinline__ unsigned f2key(float f) {
    unsigned u = __float_as_uint(f);
    return (u & 0x80000000u) ? ~u : (u | 0x80000000u);
}

__global__ __launch_bounds__(256)
void k2_kmax_tanh(const float* __restrict__ h1, float* __restrict__ p1)
{
    __shared__ unsigned keys[L1];
    __shared__ float    vals[L1];
    __shared__ int      hist[256];
    __shared__ unsigned sPrefix;
    __shared__ int      sK;
    __shared__ int      cnt;
    __shared__ int      scanGT[256];
    __shared__ int      scanEQ[256];
    const int tid = threadIdx.x;
    const size_t row = blockIdx.x;                 // 0 .. 32*768-1
    const float* src = h1 + row * L1;

    for (int i = tid; i < L1; i += 256) { float v = src[i]; vals[i] = v; keys[i] = f2key(v); }
    if (tid == 0) { sPrefix = 0u; sK = KP1; }
    __syncthreads();

    // 4-pass MSB->LSB radix select for the KP1-th largest key
    for (int p = 0; p < 4; ++p) {
        const int shift = 24 - 8 * p;
        hist[tid] = 0;                             // 256 threads == 256 bins
        __syncthreads();
        const unsigned pref = sPrefix;
        const int      k    = sK;
        for (int i = tid; i < L1; i += 256) {
            unsigned key = keys[i];
            bool match = (p == 0) || ((key >> (shift + 8)) == pref);
            if (match) atomicAdd(&hist[(key >> shift) & 0xFFu], 1);
        }
        __syncthreads();
        if (tid == 0) {
            int cum = 0, bin = 255;
            for (; bin >= 0; --bin) {
                int h = hist[bin];
                if (cum + h >= k) break;
                cum += h;
            }
            sPrefix = (pref << 8) | (unsigned)bin;
            sK = k - cum;                          // rank within selected bin
        }
        __syncthreads();
    }
    const unsigned T = sPrefix;                    // exact KP1-th largest key

    // count strictly-greater elements
    if (tid == 0) cnt = 0;
    __syncthreads();
    { int c = 0; for (int i = tid; i < L1; i += 256) c += (keys[i] > T) ? 1 : 0; atomicAdd(&cnt, c); }
    __syncthreads();
    const int nGT = cnt;
    const int nEQneed = KP1 - nGT;

    // contiguous 5-element chunks per thread (order-preserving compaction)
    const int beg = tid * 5;
    const int end = (beg + 5 < L1) ? beg + 5 : L1;
    int gl = 0, el = 0;
    for (int i = beg; i < end; ++i) { gl += (keys[i] > T) ? 1 : 0; el += (keys[i] == T) ? 1 : 0; }
    scanGT[tid] = gl; scanEQ[tid] = el;
    __syncthreads();
    if (tid == 0) {
        int ag = 0, ae = 0;
        for (int t = 0; t < 256; ++t) {
            int g = scanGT[t], e = scanEQ[t];
            scanGT[t] = ag; scanEQ[t] = ae;
            ag += g; ae += e;
        }
    }
    __syncthreads();
    int g = scanGT[tid], e = scanEQ[tid];
    float* dst = p1 + row * KP1;
    for (int i = beg; i < end; ++i) {
        unsigned kk = keys[i];
        if (kk > T) {
            int eqb = (e < nEQneed) ? e : nEQneed;
            dst[g + eqb] = tanhf(vals[i]);
            ++g;
        } else if (kk == T) {
            if (e < nEQneed) dst[g + e] = tanhf(vals[i]);
            ++e;
        }
    }
}

// ---------------------------------------------------------------------------
// Kernel 3: grouped conv2 (6->14 per group, K=5, pad=4) + fold-sum.
// One block per (b, folded group j2 in [0,64)): reads p1 channels
// [12*j2, 12*j2+12), writes h2[b, j2*14+f2, l], l in [0,519).
// ---------------------------------------------------------------------------
__global__ __launch_bounds__(256)
void k3_conv2_fold(const float* __restrict__ p1, const float* __restrict__ w2,
                   const float* __restrict__ b2, float* __restrict__ h2)
{
    __shared__ float in[12 * 523];     // 515 + 2*4 padding
    __shared__ float wl[840];          // 2 groups * 14 * 6 * 5
    const int tid = threadIdx.x;
    const int b   = blockIdx.x >> 6;
    const int j2  = blockIdx.x & 63;

    __builtin_prefetch(w2 + 840 * j2, 0, 0);   // global_prefetch_b8

    for (int t = tid; t < 840; t += 256) wl[t] = w2[840 * j2 + t];
    const float* srcBase = p1 + ((size_t)b * CH1 + 12 * j2) * KP1;
    for (int idx = tid; idx < 12 * 523; idx += 256) {
        int c = idx / 523, i = idx - c * 523;
        int s = i - 4;
        in[idx] = (s >= 0 && s < KP1) ? srcBase[(size_t)c * KP1 + s] : 0.0f;
    }
    __syncthreads();

    float* outBase = h2 + ((size_t)b * CH2 + 14 * j2) * L2;
    for (int idx = tid; idx < 14 * L2; idx += 256) {
        int f2 = idx / L2, l = idx - f2 * L2;
        float acc = b2[28 * j2 + f2] + b2[28 * j2 + 14 + f2];
        #pragma unroll
        for (int sg = 0; sg < 2; ++sg) {
            const float* wg = wl + sg * 420 + f2 * 30;
            const float* ig = in + sg * 6 * 523 + l;
            #pragma unroll
            for (int f = 0; f < 6; ++f) {
                #pragma unroll
                for (int k = 0; k < 5; ++k)
                    acc += ig[f * 523 + k] * wg[f * 5 + k];
            }
        }
        outBase[(size_t)f2 * L2 + l] = acc;
    }
}

// ---------------------------------------------------------------------------
// Kernel 4: k-max pool k=4 (order-preserving, ties lowest index) + tanh.
// One thread per (b, channel) row of 519.
// ---------------------------------------------------------------------------
__global__ __launch_bounds__(256)
void k4_kmax4_tanh(const float* __restrict__ h2, float* __restrict__ p2)
{
    const int r = blockIdx.x * 256 + threadIdx.x;   // 0..28671
    const float* src = h2 + (size_t)r * L2;
    float tv[4] = { -__builtin_inff(), -__builtin_inff(), -__builtin_inff(), -__builtin_inff() };
    int   ti[4] = { 0, 0, 0, 0 };
    for (int i = 0; i < L2; ++i) {
        float v = src[i];
        if (v > tv[3]) {                 // strict >: earlier index wins ties
            tv[3] = v; ti[3] = i;
            #pragma unroll
            for (int p = 3; p > 0; --p) {
                if (tv[p] > tv[p - 1]) { // strict >: stable for equals
                    float fv = tv[p]; tv[p] = tv[p - 1]; tv[p - 1] = fv;
                    int   fi = ti[p]; ti[p] = ti[p - 1]; ti[p - 1] = fi;
                }
            }
        }
    }
    // restore original order: sort the 4 kept entries by index ascending
    #pragma unroll
    for (int a = 0; a < 3; ++a) {
        #pragma unroll
        for (int q = 0; q < 3; ++q) {
            if (ti[q] > ti[q + 1]) {
                int   fi = ti[q]; ti[q] = ti[q + 1]; ti[q + 1] = fi;
                float fv = tv[q]; tv[q] = tv[q + 1]; tv[q + 1] = fv;
            }
        }
    }
    float* dst = p2 + (size_t)r * 4;
    #pragma unroll
    for (int q = 0; q < 4; ++q) dst[q] = tanhf(tv[q]);
}

// ---------------------------------------------------------------------------
// Kernel 5: final FC with fp32 WMMA (V_WMMA_F32_16X16X4_F32).
// out(32,1000) = A(32,3584) @ wf^T + bf. One 16x16 tile per wave.
// A/B per-lane layout: lanes 0-15 hold K=k0,k0+1; lanes 16-31 hold K=k0+2,k0+3.
// ---------------------------------------------------------------------------
__global__ __launch_bounds__(256)
void k5_wmma_fc(const float* __restrict__ A, const float* __restrict__ wf,
                const float* __restrict__ bf, float* __restrict__ out)
{
    const int lane   = threadIdx.x & 31;
    const int waveId = (blockIdx.x << 3) + (threadIdx.x >> 5);
    if (waveId >= 126) return;                 // 2 M-tiles * 63 N-tiles
    const int mt = waveId & 1;
    const int nt = waveId >> 1;
    const int m0 = mt << 4, n0 = nt << 4;
    const int hb   = lane >> 4;                // half-wave select (K pair)
    const int l15  = lane & 15;
    const int koff = hb << 1;

    const int m = m0 + l15;
    int n = n0 + l15;
    if (n > NCLS - 1) n = NCLS - 1;            // clamp loads for padded N tile

    const float* aP = A  + (size_t)m * KFLAT + koff;
    const float* bP = wf + (size_t)n * KFLAT + koff;

    v8f c = {0.f, 0.f, 0.f, 0.f, 0.f, 0.f, 0.f, 0.f};
    #pragma unroll 4
    for (int k0 = 0; k0 < KFLAT; k0 += 4) {
        v2f a  = *(const v2f*)(aP + k0);
        v2f bm = *(const v2f*)(bP + k0);
        c = __builtin_amdgcn_wmma_f32_16x16x4_f32(
            /*neg_a=*/false, a, /*neg_b=*/false, bm,
            /*c_mod=*/(short)0, c, /*reuse_a=*/false, /*reuse_b=*/false);
    }

    const int col = n0 + l15;
    if (col < NCLS) {
        #pragma unroll
        for (int r = 0; r < 8; ++r) {
            int rowi = m0 + r + (hb << 3);     // VGPR r: M=r (lo half), M=r+8 (hi half)
            out[(size_t)rowi * NCLS + col] = c[r] + bf[col];
        }
    }
}

// ---------------------------------------------------------------------------
// Launch: ws ping-pong — region A: h1 (25,313,280 f) then h2 (14,880,768 f);
//         region B: p1 (12,656,640 f) then p2 (114,688 f). Total ~152 MB.
// ---------------------------------------------------------------------------
extern "C" void kernel_launch(void* const* d_in, const int* in_sizes, int n_in,
                              void* d_out, int out_size, void* d_ws, size_t ws_size,
                              hipStream_t stream)
{
    const int*   x   = (const int*)d_in[0];
    const float* emb = (const float*)d_in[1];
    const float* w1  = (const float*)d_in[2];
    const float* b1  = (const float*)d_in[3];
    const float* w2  = (const float*)d_in[4];
    const float* b2  = (const float*)d_in[5];
    const float* wf  = (const float*)d_in[6];
    const float* bf  = (const float*)d_in[7];
    float* out = (float*)d_out;

    float* wsA = (float*)d_ws;                 // h1, later h2
    float* wsB = wsA + 25313280;               // p1, later p2

    k1_embed_conv_fold<<<BATCH * 128, 256, 0, stream>>>(x, emb, w1, b1, wsA);
    k2_kmax_tanh      <<<BATCH * CH1, 256, 0, stream>>>(wsA, wsB);
    k3_conv2_fold     <<<BATCH * 64,  256, 0, stream>>>(wsB, w2, b2, wsA);
    k4_kmax4_tanh     <<<(BATCH * CH2) / 256, 256, 0, stream>>>(wsA, wsB);
    k5_wmma_fc        <<<16, 256, 0, stream>>>(wsB, wf, bf, out);
}